// LogSlaterDeterminant_72370198937640
// MI455X (gfx1250) — compile-verified
//
#include <hip/hip_runtime.h>
#include <hip/hip_bf16.h>
#include <math.h>

// ---------------------------------------------------------------------------
// Log Slater determinant: per sample, gather 128 rows of M (256x128) selected
// by occupation vector n, LU-factor the 128x128 matrix with partial pivoting
// in LDS, trailing updates on the f32 WMMA pipe (v_wmma_f32_16x16x4_f32),
// emit complex log-det = (sum log|U_ii|, pi * sign_parity).
// ---------------------------------------------------------------------------

#define N_ORB 256
#define N_F   128
#define NB    16
#define STR   129          // LDS row stride (pad to avoid bank conflicts)
#define PI_F  3.14159265358979323846f

typedef __attribute__((ext_vector_type(2))) float v2f;
typedef __attribute__((ext_vector_type(8))) float v8f;

__global__ __launch_bounds__(256) void
logslater_lu_wmma_kernel(const int* __restrict__ nvec,
                         const float* __restrict__ M,
                         float* __restrict__ out)
{
    __shared__ float As[N_F * STR];   // 128 x 128 matrix, padded rows (~66 KB)
    __shared__ int   Ridx[N_F];       // occupied orbital indices (ascending)
    __shared__ int   pivRowS;

    const int tid  = threadIdx.x;
    const int lane = tid & 31;
    const int wid  = tid >> 5;
    const int b    = blockIdx.x;

    // ---------------- 1) compact occupied indices (wave 0) -----------------
    if (wid == 0) {
        const int* nrow = nvec + (size_t)b * N_ORB;
        int base = 0;
        #pragma unroll
        for (int g = 0; g < N_ORB / 32; ++g) {
            int o = g * 32 + lane;
            int flag = (nrow[o] != 0);
            unsigned mask = (unsigned)__ballot(flag);   // wave32: low 32 bits
            int prefix = __popc(mask & ((1u << lane) - 1u));
            if (flag && (base + prefix) < N_F) Ridx[base + prefix] = o;
            base += __popc(mask);
        }
    }
    __syncthreads();

    // ---------------- 2) gather A = M[R, :] into LDS -----------------------
    for (int idx = tid; idx < N_F * N_F; idx += 256) {
        int i = idx >> 7;
        int j = idx & (N_F - 1);
        As[i * STR + j] = M[(size_t)Ridx[i] * N_F + j];
    }
    __syncthreads();

    // ---------------- 3) blocked LU with partial pivoting ------------------
    int swaps = 0;                                // valid in wave 0 (uniform)
    const int m16  = lane & 15;
    const int half = lane >> 4;

    for (int k0 = 0; k0 < N_F; k0 += NB) {
        const int kend = k0 + NB;

        // --- panel factorization: columns k0..kend-1, unblocked ---
        for (int j = k0; j < kend; ++j) {
            if (wid == 0) {
                // pivot search over rows j..127
                float best = -1.0f; int bi = j;
                for (int i = j + lane; i < N_F; i += 32) {
                    float v = fabsf(As[i * STR + j]);
                    if (v > best) { best = v; bi = i; }
                }
                #pragma unroll
                for (int off = 16; off > 0; off >>= 1) {
                    float ov = __shfl_xor(best, off, 32);
                    int   oi = __shfl_xor(bi,  off, 32);
                    if (ov > best || (ov == best && oi < bi)) { best = ov; bi = oi; }
                }
                if (bi != j) ++swaps;             // uniform across wave 0
                if (lane == 0) pivRowS = bi;
            }
            __syncthreads();

            const int piv = pivRowS;
            if (piv != j && tid < N_F) {          // swap full rows j <-> piv
                float t0 = As[j   * STR + tid];
                As[j   * STR + tid] = As[piv * STR + tid];
                As[piv * STR + tid] = t0;
            }
            __syncthreads();

            // scale column + rank-1 update inside panel: one thread per row
            const int i = j + 1 + tid;
            if (i < N_F) {
                float pv = As[j * STR + j];
                float l  = As[i * STR + j] / pv;
                As[i * STR + j] = l;
                for (int jj = j + 1; jj < kend; ++jj)
                    As[i * STR + jj] -= l * As[j * STR + jj];
            }
            __syncthreads();
        }

        // --- triangular solve U12 = L11^{-1} * A[k0:kend, kend:] ---
        {
            const int c = kend + tid;
            if (c < N_F) {
                float x[NB];
                #pragma unroll
                for (int r = 0; r < NB; ++r) x[r] = As[(k0 + r) * STR + c];
                #pragma unroll
                for (int r = 1; r < NB; ++r) {
                    float acc = x[r];
                    #pragma unroll
                    for (int s = 0; s < NB - 1; ++s)
                        if (s < r) acc -= As[(k0 + r) * STR + (k0 + s)] * x[s];
                    x[r] = acc;
                }
                #pragma unroll
                for (int r = 1; r < NB; ++r) As[(k0 + r) * STR + c] = x[r];
            }
        }
        __syncthreads();

        // --- trailing update  A22 -= L21 * U12  on the WMMA pipe ---
        // D = (-A_frag) * B_frag + C_frag, tiles of 16x16, K = 16 (4 x k4).
        {
            const int mt = (N_F - kend) >> 4;     // tiles per dimension
            for (int t = wid; t < mt * mt; t += 8) {
                const int row0 = kend + (t / mt) * 16;
                const int col0 = kend + (t % mt) * 16;

                // C fragment (16x16 f32): VGPR v -> rows v (lanes 0-15), v+8
                v8f c;
                #pragma unroll
                for (int v = 0; v < 8; ++v)
                    c[v] = As[(row0 + v + 8 * half) * STR + col0 + m16];

                #pragma unroll
                for (int kk = 0; kk < 4; ++kk) {
                    const int kb = k0 + 4 * kk;
                    // A fragment 16x4: lane row = m16, K pair selected by half
                    v2f a;
                    a.x = -As[(row0 + m16) * STR + kb + 2 * half];
                    a.y = -As[(row0 + m16) * STR + kb + 2 * half + 1];
                    // B fragment 4x16: VGPR v -> K rows v + 2*half
                    v2f bb;
                    bb.x = As[(kb + 2 * half)     * STR + col0 + m16];
                    bb.y = As[(kb + 2 * half + 1) * STR + col0 + m16];
                    c = __builtin_amdgcn_wmma_f32_16x16x4_f32(
                            /*neg_a=*/false, a, /*neg_b=*/false, bb,
                            /*c_mod=*/(short)0, c,
                            /*reuse_a=*/false, /*reuse_b=*/false);
                }

                #pragma unroll
                for (int v = 0; v < 8; ++v)
                    As[(row0 + v + 8 * half) * STR + col0 + m16] = c[v];
            }
        }
        __syncthreads();
    }

    // ---------------- 4) log|det| and sign from U diagonal -----------------
    if (wid == 0) {
        float lsum = 0.0f; int neg = 0;
        for (int d = lane; d < N_F; d += 32) {
            float u = As[d * STR + d];
            lsum += logf(fabsf(u));
            neg  += (u < 0.0f) ? 1 : 0;
        }
        #pragma unroll
        for (int off = 16; off > 0; off >>= 1) {
            lsum += __shfl_xor(lsum, off, 32);
            neg  += __shfl_xor(neg,  off, 32);
        }
        if (lane == 0) {
            int parity = (neg + swaps) & 1;       // sign = (-1)^(swaps+negdiag)
            out[2 * b + 0] = lsum;                // Re: log|det|
            out[2 * b + 1] = parity ? PI_F : 0.0f;// Im: log(sign)
        }
    }
}

extern "C" void kernel_launch(void* const* d_in, const int* in_sizes, int n_in,
                              void* d_out, int out_size, void* d_ws, size_t ws_size,
                              hipStream_t stream)
{
    const int*   nvec = (const int*)d_in[0];     // (BATCH, 256) int32
    const float* M    = (const float*)d_in[1];   // (256, 128) float32
    float*       out  = (float*)d_out;           // (BATCH,) complex64 -> float pairs

    const int batch = in_sizes[0] / N_ORB;       // 4096

    logslater_lu_wmma_kernel<<<batch, 256, 0, stream>>>(nvec, M, out);
}